// STransformer_49890340110475
// MI455X (gfx1250) — compile-verified
//
#include <hip/hip_runtime.h>
#include <math.h>

// ---------------------------------------------------------------------------
// STransformer forward for MI455X (gfx1250, wave32, WMMA).
// fp32 reference -> V_WMMA_F32_16X16X4_F32 (chained K=4 steps) for all
// GEMM-shaped work. ~37 GFLOP fp32 vs ~350MB traffic: the N x N attention /
// GAT aggregation stages are compute-shaped -> matrix pipe.
// Round 2: batch fragment loads before WMMA chains (one s_wait_dscnt per
// group instead of per step), hoist loop-invariant fragments, and use native
// v_exp_f32 (__expf) + hoisted reciprocals in the softmax paths to cut the
// VALU expansion seen in round 1 disasm.
// ---------------------------------------------------------------------------

#define B_ 16
#define N_ 325
#define T_ 12
#define C_ 64
#define H_ 4
#define PD_ 16
#define GH_ 2
#define NT_ 21            // ceil(325/16)
#define NPAD_ 336         // 21*16
#define MTOK_ (B_*T_*N_)  // 62400  (divisible by 16 -> 3900 tiles, no row pad)
#define MTILES_ (MTOK_/16)

typedef __attribute__((ext_vector_type(2))) float v2f;
typedef __attribute__((ext_vector_type(8))) float v8f;

static __device__ __forceinline__ v8f zero8() {
  v8f z = {0.f,0.f,0.f,0.f,0.f,0.f,0.f,0.f}; return z;
}

// D = A(16x4) * B(4x16) + C   (fp32 WMMA; lane layout per CDNA5 ISA 7.12.2)
static __device__ __forceinline__ v8f wmma_f32(v2f a, v2f b, v8f c) {
  return __builtin_amdgcn_wmma_f32_16x16x4_f32(false, a, false, b, (short)0, c,
                                               false, false);
}

static __device__ __forceinline__ float fexp(float x) { return __expf(x); }
static __device__ __forceinline__ float sigm(float x) {
  return 1.0f / (1.0f + __expf(-x));
}

// ---------------------------------------------------------------------------
// DS[n,c] = sum_j adj[n,j] * W_emb[c,j] + b_emb[c]
// ---------------------------------------------------------------------------
__global__ void k_ds(const float* __restrict__ adj, const float* __restrict__ Wemb,
                     const float* __restrict__ bemb, float* __restrict__ DS) {
  int idx = blockIdx.x * blockDim.x + threadIdx.x;
  if (idx >= N_ * C_) return;
  int n = idx / C_, c = idx - n * C_;
  float acc = bemb[c];
  for (int j = 0; j < N_; ++j) acc += adj[n * N_ + j] * Wemb[c * N_ + j];
  DS[idx] = acc;
}

// ---------------------------------------------------------------------------
// qq = query + DS (stored BTNC); qh/kh/vh = per-head 16x16 projections (BTNC)
// ---------------------------------------------------------------------------
__global__ void k_qkv(const float* __restrict__ qin, const float* __restrict__ kin,
                      const float* __restrict__ vin, const float* __restrict__ DS,
                      const float* __restrict__ Wq, const float* __restrict__ Wk,
                      const float* __restrict__ Wv,
                      float* __restrict__ qq, float* __restrict__ qh,
                      float* __restrict__ kh, float* __restrict__ vh) {
  long idx = (long)blockIdx.x * blockDim.x + threadIdx.x;
  if (idx >= (long)MTOK_ * C_) return;
  int c = (int)(idx & (C_ - 1));
  long tok = idx >> 6;                 // (b,t,n)
  int n = (int)(tok % N_);
  int t = (int)((tok / N_) % T_);
  int b = (int)(tok / ((long)N_ * T_));
  long inBase = (((long)b * N_ + n) * T_ + t) * C_;   // BNTC inputs
  int hh = c >> 4, e = c & 15, dbase = hh * 16;
  qq[idx] = qin[inBase + c] + DS[n * C_ + c];
  float aq = 0.f, ak = 0.f, av = 0.f;
  for (int d = 0; d < 16; ++d) {
    float dsd = DS[n * C_ + dbase + d];
    float w_q = Wq[e * 16 + d], w_k = Wk[e * 16 + d], w_v = Wv[e * 16 + d];
    aq += (qin[inBase + dbase + d] + dsd) * w_q;
    ak += (kin[inBase + dbase + d] + dsd) * w_k;
    av += (vin[inBase + dbase + d] + dsd) * w_v;
  }
  qh[idx] = aq; kh[idx] = ak; vh[idx] = av;
}

// ---------------------------------------------------------------------------
// Attention per (b,t,head): S = qh kh^T / 8 ; softmax over QUERY axis (cols
// normalized over rows); o = A vh.  One WG per (b,t,h); q/k/v slices in LDS.
// ---------------------------------------------------------------------------
__global__ void k_attn(const float* __restrict__ qh, const float* __restrict__ kh,
                       const float* __restrict__ vh, float* __restrict__ o) {
  extern __shared__ float sm[];
  float* qs = sm;                        // NPAD_*17
  float* ks = qs + NPAD_ * 17;
  float* vs = ks + NPAD_ * 17;
  float* colm = vs + NPAD_ * 17;         // NPAD_
  float* colz = colm + NPAD_;
  float* wstAll = colz + NPAD_;          // nw*16*17

  int bt = blockIdx.x / H_, hd = blockIdx.x % H_;
  int tid = threadIdx.x, lane = tid & 31, wave = tid >> 5;
  int nw = blockDim.x >> 5;

  long base = (long)bt * N_ * C_ + hd * PD_;
  for (int i = tid; i < NPAD_ * PD_; i += blockDim.x) {
    int n = i >> 4, d = i & 15;
    float q = 0.f, k = 0.f, v = 0.f;
    if (n < N_) {
      long a = base + (long)n * C_ + d;
      q = qh[a]; k = kh[a]; v = vh[a];
    }
    qs[n * 17 + d] = q; ks[n * 17 + d] = k; vs[n * 17 + d] = v;
  }
  __syncthreads();

  int rw = lane & 15;
  int kkoff = (lane >> 4) * 2;
  int rowbase = (lane >> 4) * 8;

  // ---- Pass A: per-key-column online logsumexp over all query rows --------
  for (int kt = wave; kt < NT_; kt += nw) {
    v2f kb[4];
#pragma unroll
    for (int i = 0; i < 4; ++i) {           // k-tile fragments: qt-invariant
      int kk = i * 4 + kkoff;
      kb[i].x = ks[(kt * 16 + rw) * 17 + kk];
      kb[i].y = ks[(kt * 16 + rw) * 17 + kk + 1];
    }
    float m = -3.0e38f, z = 0.0f;
    for (int qt = 0; qt < NT_; ++qt) {
      v2f qa[4];
#pragma unroll
      for (int i = 0; i < 4; ++i) {
        int kk = i * 4 + kkoff;
        qa[i].x = qs[(qt * 16 + rw) * 17 + kk];
        qa[i].y = qs[(qt * 16 + rw) * 17 + kk + 1];
      }
      v8f acc = zero8();
#pragma unroll
      for (int i = 0; i < 4; ++i) acc = wmma_f32(qa[i], kb[i], acc);
#pragma unroll
      for (int r = 0; r < 8; ++r) {
        int q = qt * 16 + rowbase + r;
        if (q < N_) {
          float sv = acc[r] * 0.125f;
          float nm = fmaxf(m, sv);
          z = z * fexp(m - nm) + fexp(sv - nm);
          m = nm;
        }
      }
    }
    float m2 = __shfl_xor(m, 16, 32);
    float z2 = __shfl_xor(z, 16, 32);
    float nm = fmaxf(m, m2);
    z = z * fexp(m - nm) + z2 * fexp(m2 - nm);
    m = nm;
    if (lane < 16) { colm[kt * 16 + lane] = m; colz[kt * 16 + lane] = z; }
  }
  __syncthreads();

  // ---- Pass B: o = A @ vh, A built from recomputed S tiles ----------------
  float* wst = wstAll + wave * 16 * 17;
  for (int qt = wave; qt < NT_; qt += nw) {
    v2f qa[4];
#pragma unroll
    for (int i = 0; i < 4; ++i) {           // q-tile fragments: kt-invariant
      int kk = i * 4 + kkoff;
      qa[i].x = qs[(qt * 16 + rw) * 17 + kk];
      qa[i].y = qs[(qt * 16 + rw) * 17 + kk + 1];
    }
    v8f oacc = zero8();
    for (int kt = 0; kt < NT_; ++kt) {
      v2f kb[4];
#pragma unroll
      for (int i = 0; i < 4; ++i) {
        int kk = i * 4 + kkoff;
        kb[i].x = ks[(kt * 16 + rw) * 17 + kk];
        kb[i].y = ks[(kt * 16 + rw) * 17 + kk + 1];
      }
      v8f s = zero8();
#pragma unroll
      for (int i = 0; i < 4; ++i) s = wmma_f32(qa[i], kb[i], s);

      int kg = kt * 16 + rw;               // this lane's key column
      float cm = colm[kg];
      float rcz = 1.0f / colz[kg];
      bool kok = (kg < N_);
#pragma unroll
      for (int r = 0; r < 8; ++r) {
        float w = 0.0f;
        if (kok) w = fexp(s[r] * 0.125f - cm) * rcz;
        wst[(rowbase + r) * 17 + rw] = w;  // transpose via per-wave LDS stage
      }
      __asm__ volatile("s_wait_dscnt 0" ::: "memory");
      v2f wa[4], vb[4];
#pragma unroll
      for (int i = 0; i < 4; ++i) {
        int kk = i * 4 + kkoff;
        wa[i].x = wst[rw * 17 + kk];
        wa[i].y = wst[rw * 17 + kk + 1];
        vb[i].x = vs[(kt * 16 + kk) * 17 + rw];
        vb[i].y = vs[(kt * 16 + kk + 1) * 17 + rw];
      }
#pragma unroll
      for (int i = 0; i < 4; ++i) oacc = wmma_f32(wa[i], vb[i], oacc);
    }
#pragma unroll
    for (int r = 0; r < 8; ++r) {
      int q = qt * 16 + rowbase + r;
      if (q < N_) o[(long)bt * N_ * C_ + (long)q * C_ + hd * PD_ + rw] = oacc[r];
    }
  }
}

// ---------------------------------------------------------------------------
// Generic WMMA GEMM: Out = act( X[m,:] (row via b/t/n strides) @ W' + bias )
// mode 0: plain(+bias)   mode 1: ReLU   mode 2: sigmoid-gate epilogue
// ---------------------------------------------------------------------------
__global__ void k_gemm(const float* __restrict__ X, long xB, long xT, long xN,
                       const float* __restrict__ W, int wTrans, int K, int Nout,
                       const float* __restrict__ bias,
                       float* __restrict__ Out, long oBase, long oB, long oT, long oN,
                       int mode, const float* __restrict__ Fs,
                       const float* __restrict__ Us) {
  extern __shared__ float sm[];
  int Kp = K + 1;
  float* Wl = sm;                     // Nout*Kp
  float* Xall = sm + Nout * Kp;       // nw*16*Kp
  int tid = threadIdx.x, lane = tid & 31, wave = tid >> 5;
  int nw = blockDim.x >> 5;

  for (int i = tid; i < Nout * K; i += blockDim.x) {
    int n = i / K, k = i - n * K;
    Wl[n * Kp + k] = wTrans ? W[(long)k * Nout + n] : W[(long)n * K + k];
  }
  __syncthreads();

  int mTile = blockIdx.x * nw + wave;
  if (mTile >= MTILES_) return;
  float* Xl = Xall + wave * 16 * Kp;

  for (int r = 0; r < 16; ++r) {
    int m = mTile * 16 + r;
    int bb = m / (T_ * N_);
    int rem = m - bb * (T_ * N_);
    int t = rem / N_;
    int n = rem - t * N_;
    long ba = (long)bb * xB + (long)t * xT + (long)n * xN;
    __builtin_prefetch(X + ba, 0, 1);
    for (int k = lane; k < K; k += 32) Xl[r * Kp + k] = X[ba + k];
  }
  __asm__ volatile("s_wait_dscnt 0" ::: "memory");

  int col = lane & 15;
  int kkoff = (lane >> 4) * 2;
  int rowbase = (lane >> 4) * 8;

  long orow[8]; int tokidx[8];
#pragma unroll
  for (int r = 0; r < 8; ++r) {
    int m = mTile * 16 + rowbase + r;
    int bb = m / (T_ * N_);
    int rem = m - bb * (T_ * N_);
    int t = rem / N_;
    int n = rem - t * N_;
    orow[r] = oBase + (long)bb * oB + (long)t * oT + (long)n * oN;
    tokidx[r] = m;
  }

  int ngrp = K >> 4;                  // K is a multiple of 16 -> groups of 4 steps
  for (int nt = 0; nt < (Nout >> 4); ++nt) {
    int n0 = nt * 16;
    v8f acc = zero8();
    for (int g = 0; g < ngrp; ++g) {
      v2f a[4], b[4];
#pragma unroll
      for (int i = 0; i < 4; ++i) {
        int kk = (g * 4 + i) * 4 + kkoff;
        a[i].x = Xl[col * Kp + kk];        a[i].y = Xl[col * Kp + kk + 1];
        b[i].x = Wl[(n0 + col) * Kp + kk]; b[i].y = Wl[(n0 + col) * Kp + kk + 1];
      }
#pragma unroll
      for (int i = 0; i < 4; ++i) acc = wmma_f32(a[i], b[i], acc);
    }
    float bv = bias ? bias[n0 + col] : 0.0f;
#pragma unroll
    for (int r = 0; r < 8; ++r) {
      float v = acc[r] + bv;
      if (mode == 1) v = fmaxf(v, 0.0f);
      if (mode == 2) {
        long tc = (long)tokidx[r] * C_ + n0 + col;
        float g2 = sigm(v + Fs[tc]);
        float xg = Xl[(rowbase + r) * Kp + n0 + col];
        v = g2 * Us[tc] + (1.0f - g2) * xg;
      }
      Out[orow[r] + n0 + col] = v;
    }
  }
}

// ---------------------------------------------------------------------------
// Out = LayerNorm(A + Bx) * g + be   (one wave per token, C=64)
// ---------------------------------------------------------------------------
__global__ void k_ln(const float* __restrict__ A, const float* __restrict__ Bx,
                     const float* __restrict__ g, const float* __restrict__ be,
                     float* __restrict__ Out) {
  int tid = threadIdx.x, lane = tid & 31, wave = tid >> 5;
  long tok = (long)blockIdx.x * (blockDim.x >> 5) + wave;
  if (tok >= MTOK_) return;
  long base = tok * C_;
  float x0 = A[base + lane] + Bx[base + lane];
  float x1 = A[base + lane + 32] + Bx[base + lane + 32];
  float s = x0 + x1;
  for (int off = 16; off >= 1; off >>= 1) s += __shfl_xor(s, off, 32);
  float mean = s * (1.0f / C_);
  float d0 = x0 - mean, d1 = x1 - mean;
  float v = d0 * d0 + d1 * d1;
  for (int off = 16; off >= 1; off >>= 1) v += __shfl_xor(v, off, 32);
  float rstd = rsqrtf(v * (1.0f / C_) + 1e-5f);
  Out[base + lane]      = d0 * rstd * g[lane] + be[lane];
  Out[base + lane + 32] = d1 * rstd * g[lane + 32] + be[lane + 32];
}

// ---------------------------------------------------------------------------
// GAT masked softmax + aggregation per (b,t,head).
// Hf layout (B,T,GH,N,C).  layer1: out=hcat (B,T,N,2C) with ELU concat;
// else: out=hagg (B,T,GH,N,C).
// ---------------------------------------------------------------------------
__global__ void k_gat_aggr(const float* __restrict__ Hf, const float* __restrict__ as_,
                           const float* __restrict__ ad_, const float* __restrict__ adj,
                           float* __restrict__ Out, int layer1) {
  extern __shared__ float sm[];
  float* hs = sm;                 // NPAD_*65
  float* ess = hs + NPAD_ * 65;   // NPAD_
  float* eds = ess + NPAD_;
  int bt = blockIdx.x / GH_, hd = blockIdx.x % GH_;
  int tid = threadIdx.x, lane = tid & 31, wave = tid >> 5;
  int nw = blockDim.x >> 5;

  long hbase = ((long)bt * GH_ + hd) * N_ * C_;
  for (int i = tid; i < NPAD_ * C_; i += blockDim.x) {
    int n = i >> 6, c = i & 63;
    hs[n * 65 + c] = (n < N_) ? Hf[hbase + (long)n * C_ + c] : 0.0f;
  }
  __syncthreads();
  for (int n = tid; n < NPAD_; n += blockDim.x) {
    float es = 0.f, ed = 0.f;
    if (n < N_) {
      for (int c = 0; c < C_; ++c) {
        float hv = hs[n * 65 + c];
        es += hv * as_[hd * C_ + c];
        ed += hv * ad_[hd * C_ + c];
      }
    }
    ess[n] = es; eds[n] = ed;
  }
  __syncthreads();

  int rw = lane & 15;
  int kkoff = (lane >> 4) * 2;
  int rowbase = (lane >> 4) * 8;

  for (int qt = wave; qt < NT_; qt += nw) {
    int i = qt * 16 + rw;
    bool iok = (i < N_);
    float m = -3.0e38f, z = 0.0f;
    if (iok) {
      float esv = ess[i];
      for (int j = (lane >> 4); j < N_; j += 2) {
        float raw = esv + eds[j];
        float lr = raw > 0.0f ? raw : 0.2f * raw;
        float e = (adj[(long)i * N_ + j] > 0.0f) ? lr : -1.0e9f;
        float nm = fmaxf(m, e);
        z = z * fexp(m - nm) + fexp(e - nm);
        m = nm;
      }
    }
    float m2 = __shfl_xor(m, 16, 32);
    float z2 = __shfl_xor(z, 16, 32);
    float nm = fmaxf(m, m2);
    z = z * fexp(m - nm) + z2 * fexp(m2 - nm);
    m = nm;
    float rz = iok ? 1.0f / z : 0.0f;

    float esv = iok ? ess[i] : 0.0f;
    v8f acc[4] = {zero8(), zero8(), zero8(), zero8()};
    for (int kt = 0; kt < NT_; ++kt) {
      for (int s = 0; s < 4; ++s) {
        int j0 = kt * 16 + s * 4 + kkoff;
        int j1 = j0 + 1;
        float w0 = 0.0f, w1 = 0.0f;
        if (iok) {
          if (j0 < N_) {
            float raw = esv + eds[j0];
            float lr = raw > 0.0f ? raw : 0.2f * raw;
            float e = (adj[(long)i * N_ + j0] > 0.0f) ? lr : -1.0e9f;
            w0 = fexp(e - m) * rz;
          }
          if (j1 < N_) {
            float raw = esv + eds[j1];
            float lr = raw > 0.0f ? raw : 0.2f * raw;
            float e = (adj[(long)i * N_ + j1] > 0.0f) ? lr : -1.0e9f;
            w1 = fexp(e - m) * rz;
          }
        }
        v2f a; a.x = w0; a.y = w1;
        v2f bfr[4];
#pragma unroll
        for (int nt = 0; nt < 4; ++nt) {
          bfr[nt].x = hs[j0 * 65 + nt * 16 + rw];
          bfr[nt].y = hs[j1 * 65 + nt * 16 + rw];
        }
#pragma unroll
        for (int nt = 0; nt < 4; ++nt) acc[nt] = wmma_f32(a, bfr[nt], acc[nt]);
      }
    }
#pragma unroll
    for (int nt = 0; nt < 4; ++nt) {
#pragma unroll
      for (int r = 0; r < 8; ++r) {
        int ii = qt * 16 + rowbase + r;
        if (ii < N_) {
          float v = acc[nt][r];
          if (layer1) {
            v = v > 0.0f ? v : (fexp(v) - 1.0f);   // ELU
            Out[((long)bt * N_ + ii) * (2 * C_) + hd * C_ + nt * 16 + rw] = v;
          } else {
            Out[(((long)bt * GH_ + hd) * N_ + ii) * C_ + nt * 16 + rw] = v;
          }
        }
      }
    }
  }
}

// ---------------------------------------------------------------------------
// Y[b,t,n,c] = mean_h hagg[b, T-1-t, h, n, c]  (flip in t);  Q = sigmoid(Y)
// ---------------------------------------------------------------------------
__global__ void k_gat_combine(const float* __restrict__ hagg,
                              float* __restrict__ Y, float* __restrict__ Q) {
  long idx = (long)blockIdx.x * blockDim.x + threadIdx.x;
  if (idx >= (long)MTOK_ * C_) return;
  int c = (int)(idx & (C_ - 1));
  long r = idx >> 6;
  int n = (int)(r % N_);
  int t = (int)((r / N_) % T_);
  int b = (int)(r / ((long)N_ * T_));
  int ts = T_ - 1 - t;
  long s0 = ((((long)b * T_ + ts) * GH_ + 0) * N_ + n) * C_ + c;
  float v = 0.5f * (hagg[s0] + hagg[s0 + (long)N_ * C_]);
  Y[idx] = v;
  Q[idx] = sigm(v);
}

// ---------------------------------------------------------------------------
extern "C" void kernel_launch(void* const* d_in, const int* in_sizes, int n_in,
                              void* d_out, int out_size, void* d_ws, size_t ws_size,
                              hipStream_t stream) {
  (void)in_sizes; (void)n_in; (void)out_size; (void)ws_size;
  const float* query = (const float*)d_in[0];
  const float* key   = (const float*)d_in[1];
  const float* value = (const float*)d_in[2];
  const float* adj   = (const float*)d_in[3];
  const float* W_emb = (const float*)d_in[4];
  const float* b_emb = (const float*)d_in[5];
  const float* Wq = (const float*)d_in[6];
  const float* Wk = (const float*)d_in[7];
  const float* Wv = (const float*)d_in[8];
  const float* Wfc = (const float*)d_in[9];
  const float* bfc = (const float*)d_in[10];
  const float* g1 = (const float*)d_in[11];
  const float* b1 = (const float*)d_in[12];
  const float* g2 = (const float*)d_in[13];
  const float* b2 = (const float*)d_in[14];
  const float* Wff1 = (const float*)d_in[15];
  const float* bff1 = (const float*)d_in[16];
  const float* Wff2 = (const float*)d_in[17];
  const float* bff2 = (const float*)d_in[18];
  const float* Wfs = (const float*)d_in[19];
  const float* bfs = (const float*)d_in[20];
  const float* Wfg = (const float*)d_in[21];
  const float* bfg = (const float*)d_in[22];
  const float* gW1 = (const float*)d_in[23];
  const float* ga1s = (const float*)d_in[24];
  const float* ga1d = (const float*)d_in[25];
  const float* gW2 = (const float*)d_in[26];
  const float* ga2s = (const float*)d_in[27];
  const float* ga2d = (const float*)d_in[28];

  float* ws = (float*)d_ws;
  const long TOKC = (long)MTOK_ * C_;
  long p = 0;
  auto alloc = [&](long nElems) { float* r = ws + p; p += (nElems + 255) & ~255L; return r; };
  float* DS    = alloc((long)N_ * C_);
  float* qq    = alloc(TOKC);
  float* qhB   = alloc(TOKC);
  float* khB   = alloc(TOKC);
  float* vhB   = alloc(TOKC);
  float* obuf  = alloc(TOKC);
  float* o2    = alloc(TOKC);
  float* Ms    = alloc(TOKC);
  float* ff1   = alloc(TOKC * 4);
  float* ff2   = alloc(TOKC);
  float* Us    = alloc(TOKC);
  float* Fs    = alloc(TOKC);
  float* hfeat = alloc(TOKC * GH_);
  float* hcat  = alloc(TOKC * 2);
  float* hagg  = alloc(TOKC * GH_);
  float* Y1    = alloc(TOKC);
  float* Q1    = alloc(TOKC);
  float* Y2    = alloc(TOKC);
  float* Q2    = alloc(TOKC);

  const long SB = (long)T_ * N_ * C_, ST = (long)N_ * C_, SN = C_;   // BTNC
  const size_t attnSmem = (size_t)(3 * NPAD_ * 17 + 2 * NPAD_ + 8 * 16 * 17) * sizeof(float);
  const size_t aggrSmem = (size_t)(NPAD_ * 65 + 2 * NPAD_) * sizeof(float);

  auto gemm = [&](const float* X, long xB, long xT, long xN,
                  const float* W, int wT, int K, int Nout, const float* bias,
                  float* Out, long oBase, long oB, long oT, long oN,
                  int mode, const float* Fsp, const float* Usp) {
    const int nw = 8;
    size_t smem = (size_t)(Nout * (K + 1) + nw * 16 * (K + 1)) * sizeof(float);
    int grid = (MTILES_ + nw - 1) / nw;
    k_gemm<<<grid, nw * 32, smem, stream>>>(X, xB, xT, xN, W, wT, K, Nout, bias,
                                            Out, oBase, oB, oT, oN, mode, Fsp, Usp);
  };

  // ---- attention branch ----------------------------------------------------
  k_ds<<<(N_ * C_ + 255) / 256, 256, 0, stream>>>(adj, W_emb, b_emb, DS);
  k_qkv<<<(int)((TOKC + 255) / 256), 256, 0, stream>>>(query, key, value, DS,
                                                       Wq, Wk, Wv, qq, qhB, khB, vhB);
  k_attn<<<B_ * T_ * H_, 256, attnSmem, stream>>>(qhB, khB, vhB, obuf);
  gemm(obuf, SB, ST, SN, Wfc, 0, C_, C_, bfc, o2, 0, SB, ST, SN, 0, nullptr, nullptr);
  k_ln<<<(MTOK_ + 7) / 8, 256, 0, stream>>>(o2, qq, g1, b1, Ms);
  gemm(Ms, SB, ST, SN, Wff1, 0, C_, 4 * C_, bff1, ff1,
       0, (long)T_ * N_ * 4 * C_, (long)N_ * 4 * C_, 4 * C_, 1, nullptr, nullptr);
  gemm(ff1, (long)T_ * N_ * 4 * C_, (long)N_ * 4 * C_, (long)4 * C_, Wff2, 0, 4 * C_, C_,
       bff2, ff2, 0, SB, ST, SN, 0, nullptr, nullptr);
  k_ln<<<(MTOK_ + 7) / 8, 256, 0, stream>>>(ff2, Ms, g2, b2, Us);
  gemm(Us, SB, ST, SN, Wfs, 0, C_, C_, bfs, Fs, 0, SB, ST, SN, 0, nullptr, nullptr);

  // ---- GAT branch (2 passes) ----------------------------------------------
  const long hOB = (long)T_ * GH_ * N_ * C_, hOT = (long)GH_ * N_ * C_;
  auto gatpass = [&](const float* Xin, long xB, long xT, long xN,
                     float* Yout, float* Qout) {
    for (int h = 0; h < GH_; ++h)
      gemm(Xin, xB, xT, xN, gW1 + (long)h * C_ * C_, 1, C_, C_, nullptr,
           hfeat, (long)h * N_ * C_, hOB, hOT, C_, 0, nullptr, nullptr);
    k_gat_aggr<<<B_ * T_ * GH_, 256, aggrSmem, stream>>>(hfeat, ga1s, ga1d, adj, hcat, 1);
    for (int h = 0; h < GH_; ++h)
      gemm(hcat, (long)T_ * N_ * 2 * C_, (long)N_ * 2 * C_, (long)2 * C_,
           gW2 + (long)h * 2 * C_ * C_, 1, 2 * C_, C_, nullptr,
           hfeat, (long)h * N_ * C_, hOB, hOT, C_, 0, nullptr, nullptr);
    k_gat_aggr<<<B_ * T_ * GH_, 256, aggrSmem, stream>>>(hfeat, ga2s, ga2d, adj, hagg, 0);
    k_gat_combine<<<(int)((TOKC + 255) / 256), 256, 0, stream>>>(hagg, Yout, Qout);
  };
  gatpass(query, (long)N_ * T_ * C_, C_, (long)T_ * C_, Y1, Q1);  // pass 1: x = raw query (BNTC)
  gatpass(Q1, SB, ST, SN, Y2, Q2);                                 // pass 2: x = sigmoid(Y1)

  // ---- final gating: out[l] = g*Us + (1-g)*XG, written BNTC into d_out ----
  const float* XGs[4] = {Q1, Y1, Q2, Y2};
  for (int l = 0; l < 4; ++l)
    gemm(XGs[l], SB, ST, SN, Wfg, 0, C_, C_, bfg, (float*)d_out,
         (long)l * B_ * N_ * T_ * C_, (long)N_ * T_ * C_, C_, (long)T_ * C_,
         2, Fs, Us);
}